// SS2NER_35948876268308
// MI455X (gfx1250) — compile-verified
//
#include <hip/hip_runtime.h>
#include <cfloat>

typedef __attribute__((ext_vector_type(4))) unsigned v4u;
typedef __attribute__((ext_vector_type(8))) unsigned v8u;

// Problem constants (from reference setup_inputs)
#define BB   4      // batch
#define LP   256    // wordpieces
#define LW   200    // words
#define DD   768    // hidden dim
#define DT   64     // d-tile staged in LDS (LP*DT*4 = 64KB)
#define NDT  (DD / DT)   // 12 d-tiles
#define WPG  50     // words per block
#define NWG  (LW / WPG)  // 4 word groups

// ---------------- Pass 1: global min of bert_emb ----------------

__global__ __launch_bounds__(256) void ss2ner_min_partial(
    const float* __restrict__ x, int n, float* __restrict__ part) {
  __shared__ float red[256];
  const int tid = threadIdx.x;
  float m = FLT_MAX;
  for (int i = blockIdx.x * 256 + tid; i < n; i += gridDim.x * 256)
    m = fminf(m, x[i]);
  red[tid] = m;
  __syncthreads();
  for (int s = 128; s > 0; s >>= 1) {
    if (tid < s) red[tid] = fminf(red[tid], red[tid + s]);
    __syncthreads();
  }
  if (tid == 0) part[blockIdx.x] = red[0];
}

__global__ __launch_bounds__(256) void ss2ner_min_final(
    const float* __restrict__ part, float* __restrict__ outv) {
  __shared__ float red[256];
  const int tid = threadIdx.x;
  red[tid] = part[tid];
  __syncthreads();
  for (int s = 128; s > 0; s >>= 1) {
    if (tid < s) red[tid] = fminf(red[tid], red[tid + s]);
    __syncthreads();
  }
  if (tid == 0) outv[0] = red[0];
}

// ---------------- Pass 2: masked max-pool words ----------------
//
// Block = (d-tile, word-group, batch). One wave programs the Tensor Data
// Mover with a 2D D# (256 rows x 64 f32, row stride 768 f32) to DMA the
// 64 KB emb tile into LDS (TENSORcnt-tracked). Meanwhile all waves pack
// pieces2word rows into wave-uniform LDS bitmasks. After s_wait_tensorcnt
// + barrier, each wave scalar-iterates set bits (readfirstlane + ctz),
// doing ds_load_b32 + v_max_num_f32 only for pieces actually present.

__global__ __launch_bounds__(256) void ss2ner_word_max(
    const float* __restrict__ emb, const int* __restrict__ mask,
    const float* __restrict__ minv_p, float* __restrict__ out) {
  extern __shared__ char smem[];           // LP*DT floats = 64 KB tile
  float* tile = (float*)smem;
  __shared__ unsigned mb[WPG * (LP / 32)]; // 50 words x 8 u32 bitmask

  const int tid   = threadIdx.x;
  const int dtile = blockIdx.x;   // 0..11
  const int wg    = blockIdx.y;   // 0..3
  const int b     = blockIdx.z;   // 0..3
  const int d0    = dtile * DT;

  // Scalarized wave id so the guard is a true s_cbranch (TDM ignores EXEC;
  // without a scalar branch every wave would re-issue the DMA).
  const int lane0_tid = __builtin_amdgcn_readfirstlane((int)tid);

  if (lane0_tid == 0) {  // wave 0 programs the Tensor Data Mover
    const unsigned lds_base = (unsigned)(size_t)(void*)tile;  // LDS byte addr
    const unsigned long long ga =
        (unsigned long long)(size_t)emb +
        ((unsigned long long)b * LP * DD + (unsigned long long)d0) * 4ull;

    v4u g0;
    g0.x = 1u;                                  // count=1 (valid), user mode
    g0.y = lds_base;                            // lds_addr
    g0.z = (unsigned)ga;                        // global_addr[31:0]
    g0.w = (unsigned)(ga >> 32) | (2u << 30);   // global_addr[56:32] | type=2

    v8u g1;
    g1[0] = 2u << 16;             // workgroup_mask=0 | data_size=4B | no pad
    g1[1] = (unsigned)DD << 16;   // barrier_addr=0 | tensor_dim0[15:0]=768
    g1[2] = (unsigned)LP << 16;   // tensor_dim0[31:16]=0 | tensor_dim1[15:0]=256
    g1[3] = (unsigned)DT << 16;   // tensor_dim1[31:16]=0 | tile_dim0=64
    g1[4] = (unsigned)LP;         // tile_dim1=256 | tile_dim2=0 (2D)
    g1[5] = (unsigned)DD;         // tensor_dim0_stride[31:0]=768
    g1[6] = 0u;                   // stride0[47:32]=0 | stride1[15:0]=0
    g1[7] = 0u;                   // tensor_dim1_stride[47:16]=0

    // 2-group form: VADDR2/VADDR3 = NULL (tensor <= 2D)
    asm volatile("tensor_load_to_lds %0, %1" :: "s"(g0), "s"(g1) : "memory");
  }

  // ---- pack mask bits while the TDM DMA is in flight ----
  for (int j = tid; j < WPG * 8; j += 256) {
    const int wl = j >> 3;    // word within group
    const int c  = j & 7;     // 32-piece chunk
    const int* mrow =
        mask + ((size_t)b * LW + (size_t)(wg * WPG + wl)) * LP + c * 32;
    unsigned bits = 0u;
#pragma unroll
    for (int k = 0; k < 32; ++k)
      bits |= (mrow[k] != 0 ? 1u : 0u) << k;
    mb[j] = bits;
  }

  if (lane0_tid == 0)
    __builtin_amdgcn_s_wait_tensorcnt(0);  // tile resident in LDS
  __syncthreads();                         // publish tile + bitmasks

  const float minv = minv_p[0];
  const int dcol = tid & (DT - 1);  // 0..63 : waves cover 32 contiguous dcols
  const int wsub = tid >> 6;        // 0..3  : wave-uniform word subgroup

  for (int wl = wsub; wl < WPG; wl += 4) {
    float acc = minv;               // all-masked word -> min_value (matches ref)
#pragma unroll 1
    for (int c = 0; c < 8; ++c) {
      // bits identical across the wave -> hoist to SGPR, scalar bit loop
      unsigned bits =
          (unsigned)__builtin_amdgcn_readfirstlane((int)mb[wl * 8 + c]);
      while (bits) {
        const int k = __builtin_ctz(bits);
        bits &= bits - 1u;
        acc = fmaxf(acc, tile[(c * 32 + k) * DT + dcol]);
      }
    }
    out[((size_t)b * LW + (size_t)(wg * WPG + wl)) * DD + d0 + dcol] = acc;
  }
}

// ---------------- launcher ----------------

extern "C" void kernel_launch(void* const* d_in, const int* in_sizes, int n_in,
                              void* d_out, int out_size, void* d_ws, size_t ws_size,
                              hipStream_t stream) {
  const float* emb  = (const float*)d_in[0];  // bert_emb f32 [4,256,768]
  const int*   mask = (const int*)d_in[1];    // pieces2word i32 [4,200,256]
  float* ws  = (float*)d_ws;                  // [0..255] partials, [256] min
  float* out = (float*)d_out;                 // [4,200,768] f32

  const int n = in_sizes[0];                  // 786432

  ss2ner_min_partial<<<256, 256, 0, stream>>>(emb, n, ws);
  ss2ner_min_final<<<1, 256, 0, stream>>>(ws, ws + 256);

  dim3 grid(NDT, NWG, BB);                    // 12 x 4 x 4 = 192 blocks
  const size_t smem = (size_t)LP * DT * sizeof(float);  // 64 KB dynamic LDS
  ss2ner_word_max<<<grid, 256, smem, stream>>>(emb, mask, ws + 256, out);
}